// NetWrapper_9552007266882
// MI455X (gfx1250) — compile-verified
//
#include <hip/hip_runtime.h>

// CDNA5 wave32 ext-vector operand types for V_WMMA_F32_16X16X4_F32:
// A = 16x4 f32  -> 64 elems / 32 lanes = 2 VGPRs
// B = 4x16 f32  -> 2 VGPRs
// C/D = 16x16 f32 -> 8 VGPRs
typedef __attribute__((ext_vector_type(2))) float v2f;
typedef __attribute__((ext_vector_type(8))) float v8f;

#define BS   4
#define NC   128      // C (input channels)
#define HID  256
#define NOUT 64
#define EPS  1e-5f

// LDS row strides padded so column walks (stride mod 64 banks == 4) hit
// 16/32 distinct banks per wave-wide operand load.
#define FSTR (NC + 4)    // 132
#define HSTR (HID + 4)   // 260

__global__ __launch_bounds__(512) void pyramid_mlp_wmma(
    const float* __restrict__ x0, const float* __restrict__ x1,
    const float* __restrict__ x2, const float* __restrict__ x3,
    const float* __restrict__ x4,
    const int*   __restrict__ p,
    const float* __restrict__ w1, const float* __restrict__ b1,
    const float* __restrict__ gamma, const float* __restrict__ beta,
    const float* __restrict__ w2, const float* __restrict__ b2,
    float* __restrict__ out)
{
    __shared__ float sF[16 * FSTR];   // gathered feats, padded to M=16 (rows 4..15 = 0)
    __shared__ float sH[16 * HSTR];   // h = feats @ W1^T + b1
    __shared__ float sR[16 * HSTR];   // relu(bn(h)), rows 4..15 = 0

    const int k    = blockIdx.x;        // pyramid level 0..4
    const int tid  = threadIdx.x;       // 0..511  (16 wave32 waves)
    const int lane = tid & 31;
    const int wave = tid >> 5;

    const float* xk = (k == 0) ? x0 : (k == 1) ? x1 : (k == 2) ? x2 : (k == 3) ? x3 : x4;
    const int side = 128 >> (k + 1);
    const int s2   = side * side;
    const int s3   = s2 * side;

    const float* w1k = w1 + (size_t)k * HID * NC;   // (HID, C) row-major
    const float* w2k = w2 + (size_t)k * NOUT * HID; // (OUT, HID) row-major

    // ---- Phase 0: zero the padded A tile; prefetch weight blocks ----
    for (int i = tid; i < 16 * FSTR; i += 512) sF[i] = 0.0f;
    {   // pull w1/w2 toward the caches while we gather (global_prefetch_b8)
        const char* pw1 = (const char*)w1k;
        const char* pw2 = (const char*)w2k;
        for (size_t off = (size_t)tid * 256; off < (size_t)HID * NC * 4; off += 512 * 256)
            __builtin_prefetch(pw1 + off, 0, 0);
        for (size_t off = (size_t)tid * 256; off < (size_t)NOUT * HID * 4; off += 512 * 256)
            __builtin_prefetch(pw2 + off, 0, 0);
    }
    __syncthreads();

    // ---- Phase 1: gather feats[b][c] = x_k[b, c, q0, q1, q2], q = p[b] >> (k+1) ----
    {
        const int b = tid >> 7;                 // 0..3
        const int c = tid & 127;                // 0..127
        const int q0 = p[b * 3 + 0] >> (k + 1);
        const int q1 = p[b * 3 + 1] >> (k + 1);
        const int q2 = p[b * 3 + 2] >> (k + 1);
        sF[b * FSTR + c] =
            xk[(size_t)(b * NC + c) * s3 + (size_t)q0 * s2 + q1 * side + q2];
    }
    __syncthreads();

    // ---- Phase 2: GEMM1 h = feats @ W1^T, one 16-wide HID tile per wave ----
    {
        const int n0 = wave * 16;               // HID column base for this wave
        const int m  = lane & 15;               // A-row (lanes 0..15 / 16..31 both M=0..15)
        const int kh = (lane >> 4) << 1;        // K sub-offset: 0 (lo half) or 2 (hi half)
        v8f acc = {};
        #pragma unroll
        for (int kk = 0; kk < NC; kk += 4) {
            v2f a, b;
            a.x = sF[m * FSTR + kk + kh];       // A(M=m, K=kk+kh / +1)
            a.y = sF[m * FSTR + kk + kh + 1];
            b.x = w1k[(size_t)(n0 + m) * NC + kk + kh];     // B(K, N=n0+m) = w1[n][c]
            b.y = w1k[(size_t)(n0 + m) * NC + kk + kh + 1];
            acc = __builtin_amdgcn_wmma_f32_16x16x4_f32(
                      false, a, false, b, (short)0, acc, false, false);
        }
        // D layout: VGPR r -> M = r + (lane<16 ? 0 : 8), N = n0 + (lane & 15)
        const int mrow = (lane >> 4) << 3;      // 0 or 8
        const int n    = n0 + m;
        const float bias = b1[k * HID + n];
        #pragma unroll
        for (int r = 0; r < 8; ++r)
            sH[(mrow + r) * HSTR + n] = acc[r] + bias;
    }
    __syncthreads();

    // ---- Phase 3: BatchNorm (train-mode stats over BS=4, biased var) + ReLU ----
    if (tid < HID) {
        const int n = tid;
        const float h0 = sH[0 * HSTR + n], h1 = sH[1 * HSTR + n];
        const float h2 = sH[2 * HSTR + n], h3 = sH[3 * HSTR + n];
        const float mean = 0.25f * (h0 + h1 + h2 + h3);
        const float d0 = h0 - mean, d1 = h1 - mean, d2 = h2 - mean, d3 = h3 - mean;
        const float var = 0.25f * (d0 * d0 + d1 * d1 + d2 * d2 + d3 * d3);
        const float scl = gamma[k * HID + n] / sqrtf(var + EPS);
        const float sh  = beta[k * HID + n];
        sR[0 * HSTR + n] = fmaxf(d0 * scl + sh, 0.0f);
        sR[1 * HSTR + n] = fmaxf(d1 * scl + sh, 0.0f);
        sR[2 * HSTR + n] = fmaxf(d2 * scl + sh, 0.0f);
        sR[3 * HSTR + n] = fmaxf(d3 * scl + sh, 0.0f);
        #pragma unroll
        for (int r = 4; r < 16; ++r) sR[r * HSTR + n] = 0.0f;  // keep A padding zero
    }
    __syncthreads();

    // ---- Phase 4: GEMM2 out = relu @ W2^T + b2; 4 waves, 16 OUT cols each ----
    if (wave < 4) {                              // wave-uniform: EXEC stays all-1
        const int n0 = wave * 16;                // OUT column base
        const int m  = lane & 15;
        const int kh = (lane >> 4) << 1;
        v8f acc = {};
        #pragma unroll 8
        for (int kk = 0; kk < HID; kk += 4) {
            v2f a, b;
            a.x = sR[m * HSTR + kk + kh];
            a.y = sR[m * HSTR + kk + kh + 1];
            b.x = w2k[(size_t)(n0 + m) * HID + kk + kh];    // B(K=c, N=o) = w2[o][c]
            b.y = w2k[(size_t)(n0 + m) * HID + kk + kh + 1];
            acc = __builtin_amdgcn_wmma_f32_16x16x4_f32(
                      false, a, false, b, (short)0, acc, false, false);
        }
        // valid outputs: M = 0..3 -> VGPRs 0..3 of lanes 0..15
        if (lane < 16) {
            const int o = n0 + m;
            const float bias = b2[k * NOUT + o];
            #pragma unroll
            for (int r = 0; r < BS; ++r)
                out[(size_t)k * BS * NOUT + r * NOUT + o] = acc[r] + bias;
        }
    }
}

extern "C" void kernel_launch(void* const* d_in, const int* in_sizes, int n_in,
                              void* d_out, int out_size, void* d_ws, size_t ws_size,
                              hipStream_t stream) {
    const float* x0    = (const float*)d_in[0];
    const float* x1    = (const float*)d_in[1];
    const float* x2    = (const float*)d_in[2];
    const float* x3    = (const float*)d_in[3];
    const float* x4    = (const float*)d_in[4];
    const int*   p     = (const int*)  d_in[5];
    const float* w1    = (const float*)d_in[6];
    const float* b1    = (const float*)d_in[7];
    const float* gamma = (const float*)d_in[8];
    const float* beta  = (const float*)d_in[9];
    const float* w2    = (const float*)d_in[10];
    const float* b2    = (const float*)d_in[11];
    float* out = (float*)d_out;   // (5, 4, 64)

    pyramid_mlp_wmma<<<5, 512, 0, stream>>>(x0, x1, x2, x3, x4, p,
                                            w1, b1, gamma, beta, w2, b2, out);
}